// GNN_62508954026537
// MI455X (gfx1250) — compile-verified
//
#include <hip/hip_runtime.h>

// MI455X / gfx1250: wave32, WMMA (V_WMMA_F32_16X16X4_F32 for fp32 GEMM),
// async global->LDS (GLOBAL_LOAD_ASYNC_TO_LDS_B128, ASYNCcnt) for staging.
typedef __attribute__((ext_vector_type(2))) float v2f;
typedef __attribute__((ext_vector_type(8))) float v8f;

#define GNN_B    8192
#define GNN_S    50
#define GNN_N    100000
#define GNN_D    128
#define A_STRIDE 132   // 128 + 4 floats pad -> conflict-free LDS A reads; 528B row, 16B aligned

// ------------------------------------------------------------------------
// Kernel A: T = relu(feat @ W)   feat:[N,128]  W:[128,128]  T:[N,128]
// 8 waves per WG, each owns a 16-col block; W fragments resident in VGPRs.
// A tiles double-buffered in LDS via async global->LDS DMA so the next
// tile's fill overlaps the current tile's 32-WMMA chain.
// ------------------------------------------------------------------------
__global__ __launch_bounds__(256) void gnn_transform_relu(
    const float* __restrict__ feat, const float* __restrict__ W,
    float* __restrict__ T, int nrows)
{
  __shared__ float Alds[2 * 16 * A_STRIDE];
  const int lane = threadIdx.x & 31;
  const int wave = threadIdx.x >> 5;
  const int col0 = wave * 16;
  const int half = lane >> 4;   // lanes 16-31 hold K+2,K+3 per ISA layout
  const int l16  = lane & 15;

  // B fragments: W[k, col0+l16]; 32 K-steps of 4
  v2f bf[32];
#pragma unroll
  for (int kk = 0; kk < 32; ++kk) {
    const int k0 = kk * 4 + half * 2;
    bf[kk].x = W[(k0 + 0) * GNN_D + col0 + l16];
    bf[kk].y = W[(k0 + 1) * GNN_D + col0 + l16];
  }

  // Per-thread async staging addresses: 512 16B chunks per tile, 2 per thread.
  const int e0 = threadIdx.x;          // chunk id 0..255
  const int e1 = threadIdx.x + 256;    // chunk id 256..511
  const unsigned ldsBase  = (unsigned)(uintptr_t)(&Alds[0]); // LDS aperture: low 32 bits = LDS byte addr
  const unsigned bufBytes = 16u * A_STRIDE * 4u;
  const unsigned ldsOff0  = ((unsigned)((e0 >> 5) * A_STRIDE + (e0 & 31) * 4)) * 4u;
  const unsigned ldsOff1  = ((unsigned)((e1 >> 5) * A_STRIDE + (e1 & 31) * 4)) * 4u;

  auto issue_tile = [&](int t, int parity) {
    const char* src = (const char*)(feat + (size_t)t * 16 * GNN_D);
    const void* g0 = (const void*)(src + (size_t)e0 * 16);
    const void* g1 = (const void*)(src + (size_t)e1 * 16);
    unsigned lds0 = ldsBase + (unsigned)parity * bufBytes + ldsOff0;
    unsigned lds1 = ldsBase + (unsigned)parity * bufBytes + ldsOff1;
    asm volatile("global_load_async_to_lds_b128 %0, %1, off"
                 :: "v"(lds0), "v"(g0) : "memory");
    asm volatile("global_load_async_to_lds_b128 %0, %1, off"
                 :: "v"(lds1), "v"(g1) : "memory");
  };

  const int ntiles = nrows >> 4;   // 100000/16 = 6250 exact
  int tile = blockIdx.x;
  int parity = 0;
  if (tile < ntiles) issue_tile(tile, 0);

  for (; tile < ntiles; tile += gridDim.x) {
    asm volatile("s_wait_asynccnt 0x0" ::: "memory");  // this wave's DMA done
    __syncthreads();                                    // all waves' DMA done

    const int tnext = tile + gridDim.x;
    if (tnext < ntiles) issue_tile(tnext, parity ^ 1);  // overlap with WMMAs below

    const float* Ab = &Alds[parity * 16 * A_STRIDE];
    v8f acc = {0.f, 0.f, 0.f, 0.f, 0.f, 0.f, 0.f, 0.f};
#pragma unroll
    for (int kk = 0; kk < 32; ++kk) {
      const int k0 = kk * 4 + half * 2;
      v2f a;
      a.x = Ab[l16 * A_STRIDE + k0];
      a.y = Ab[l16 * A_STRIDE + k0 + 1];
      acc = __builtin_amdgcn_wmma_f32_16x16x4_f32(false, a, false, bf[kk],
                                                  (short)0, acc, false, false);
    }
    // D layout: acc[v] -> row v (+8 for upper lane half), col l16
    float* dstT = T + ((size_t)tile * 16) * GNN_D + col0 + l16;
#pragma unroll
    for (int v = 0; v < 8; ++v) {
      const int r = v + half * 8;
      float x = acc[v];
      dstT[(size_t)r * GNN_D] = x > 0.f ? x : 0.f;
    }
    __syncthreads();   // free buf[parity] for the DMA issued two iterations ahead
    parity ^= 1;
  }
}

// ------------------------------------------------------------------------
// Kernel B: per batch row b (one 128-thread block, lane-per-feature d):
//   Hu[b] = l2n(Tu[u_idx[b]]);  Hi[b] = l2n(Tv[v_idx[b]])          (row l2n)
//   Mv[b,d] = (1/S) * rsqrt(max(sum_s Tv[vs,d]^2, eps)) * sum_s Tv[vs,d]*i_ew[val_s,d]
//   Mu[b,d] analogous with u-side supports.
// Gathered 512B rows are coalesced and hit L2 (102 MB tables < 192 MB L2).
// ------------------------------------------------------------------------
__global__ __launch_bounds__(128) void gnn_support_agg(
    const int* __restrict__ u_idx, const int* __restrict__ v_idx,
    const int* __restrict__ u_sup, const int* __restrict__ v_sup,
    const int* __restrict__ item_val, const int* __restrict__ user_val,
    const float* __restrict__ Tu, const float* __restrict__ Tv,
    const float* __restrict__ i_ew, const float* __restrict__ u_ew,
    float* __restrict__ Hu, float* __restrict__ Hi,
    float* __restrict__ Mv, float* __restrict__ Mu)
{
  __shared__ float red[128];
  __shared__ int sidx[GNN_S];
  __shared__ int sval[GNN_S];
  const int b = blockIdx.x;
  const int d = threadIdx.x;

  const float su = Tu[(size_t)u_idx[b] * GNN_D + d];
  const float si = Tv[(size_t)v_idx[b] * GNN_D + d];

  red[d] = su * su; __syncthreads();
  for (int off = 64; off > 0; off >>= 1) { if (d < off) red[d] += red[d + off]; __syncthreads(); }
  const float nsu = red[0]; __syncthreads();
  red[d] = si * si; __syncthreads();
  for (int off = 64; off > 0; off >>= 1) { if (d < off) red[d] += red[d + off]; __syncthreads(); }
  const float nsi = red[0]; __syncthreads();

  Hu[(size_t)b * GNN_D + d] = su * rsqrtf(fmaxf(nsu, 1e-12f));
  Hi[(size_t)b * GNN_D + d] = si * rsqrtf(fmaxf(nsi, 1e-12f));

  // v-side supports feed the USER path (i_edge_weights[item_support_val])
  if (d < GNN_S) { sidx[d] = v_sup[(size_t)b * GNN_S + d];
                   sval[d] = item_val[(size_t)b * GNN_S + d]; }
  __syncthreads();
  float ssq = 0.f, wsum = 0.f;
  for (int s = 0; s < GNN_S; ++s) {
    const float x = Tv[(size_t)sidx[s] * GNN_D + d];
    ssq  += x * x;
    wsum += x * i_ew[sval[s] * GNN_D + d];
  }
  Mv[(size_t)b * GNN_D + d] = wsum * rsqrtf(fmaxf(ssq, 1e-12f)) * (1.f / GNN_S);
  __syncthreads();

  // u-side supports feed the ITEM path (u_edge_weights[user_support_val])
  if (d < GNN_S) { sidx[d] = u_sup[(size_t)b * GNN_S + d];
                   sval[d] = user_val[(size_t)b * GNN_S + d]; }
  __syncthreads();
  ssq = 0.f; wsum = 0.f;
  for (int s = 0; s < GNN_S; ++s) {
    const float x = Tu[(size_t)sidx[s] * GNN_D + d];
    ssq  += x * x;
    wsum += x * u_ew[sval[s] * GNN_D + d];
  }
  Mu[(size_t)b * GNN_D + d] = wsum * rsqrtf(fmaxf(ssq, 1e-12f)) * (1.f / GNN_S);
}

// ------------------------------------------------------------------------
// Kernel C: V = l2n(relu(Hself @ Wself + Mneigh @ Wneigh + bias), axis=1)
// Same WMMA tiling; both weight matrices' fragments resident in VGPRs
// (accumulator carries straight through both K=128 passes).
// ------------------------------------------------------------------------
__global__ __launch_bounds__(256) void gnn_convolve_norm(
    const float* __restrict__ Hself, const float* __restrict__ Mneigh,
    const float* __restrict__ Wself, const float* __restrict__ Wneigh,
    const float* __restrict__ bias, float* __restrict__ Vout)
{
  __shared__ float Slds[16 * A_STRIDE];
  __shared__ float Nlds[16 * A_STRIDE];
  __shared__ float Vt[16 * A_STRIDE];
  __shared__ float rrow[16];
  const int lane = threadIdx.x & 31;
  const int wave = threadIdx.x >> 5;
  const int col0 = wave * 16;
  const int half = lane >> 4;
  const int l16  = lane & 15;

  v2f bs[32], bn[32];
#pragma unroll
  for (int kk = 0; kk < 32; ++kk) {
    const int k0 = kk * 4 + half * 2;
    bs[kk].x = Wself [(k0 + 0) * GNN_D + col0 + l16];
    bs[kk].y = Wself [(k0 + 1) * GNN_D + col0 + l16];
    bn[kk].x = Wneigh[(k0 + 0) * GNN_D + col0 + l16];
    bn[kk].y = Wneigh[(k0 + 1) * GNN_D + col0 + l16];
  }
  const float bcol = bias[col0 + l16];

  const int ntiles = GNN_B >> 4;   // 512
  for (int tile = blockIdx.x; tile < ntiles; tile += gridDim.x) {
    const float4* s0 = (const float4*)(Hself  + (size_t)tile * 16 * GNN_D);
    const float4* s1 = (const float4*)(Mneigh + (size_t)tile * 16 * GNN_D);
#pragma unroll
    for (int i = 0; i < 2; ++i) {
      const int e = threadIdx.x + i * 256;
      const int r = e >> 5, k4 = e & 31;
      float4 a = s0[e]; float4 c = s1[e];
      float* da = &Slds[r * A_STRIDE + k4 * 4];
      float* dc = &Nlds[r * A_STRIDE + k4 * 4];
      da[0]=a.x; da[1]=a.y; da[2]=a.z; da[3]=a.w;
      dc[0]=c.x; dc[1]=c.y; dc[2]=c.z; dc[3]=c.w;
    }
    __syncthreads();

    v8f acc = {0.f, 0.f, 0.f, 0.f, 0.f, 0.f, 0.f, 0.f};
#pragma unroll
    for (int kk = 0; kk < 32; ++kk) {
      const int k0 = kk * 4 + half * 2;
      v2f a;
      a.x = Slds[l16 * A_STRIDE + k0];
      a.y = Slds[l16 * A_STRIDE + k0 + 1];
      acc = __builtin_amdgcn_wmma_f32_16x16x4_f32(false, a, false, bs[kk],
                                                  (short)0, acc, false, false);
    }
#pragma unroll
    for (int kk = 0; kk < 32; ++kk) {
      const int k0 = kk * 4 + half * 2;
      v2f a;
      a.x = Nlds[l16 * A_STRIDE + k0];
      a.y = Nlds[l16 * A_STRIDE + k0 + 1];
      acc = __builtin_amdgcn_wmma_f32_16x16x4_f32(false, a, false, bn[kk],
                                                  (short)0, acc, false, false);
    }

    // bias + relu into LDS tile, then row-wise l2 normalize
#pragma unroll
    for (int v = 0; v < 8; ++v) {
      const int r = v + half * 8;
      float x = acc[v] + bcol;
      Vt[r * A_STRIDE + col0 + l16] = x > 0.f ? x : 0.f;
    }
    __syncthreads();
    if (threadIdx.x < 16) {
      float ssq = 0.f;
      const float* row = &Vt[threadIdx.x * A_STRIDE];
      for (int c = 0; c < GNN_D; ++c) ssq += row[c] * row[c];
      rrow[threadIdx.x] = rsqrtf(fmaxf(ssq, 1e-12f));
    }
    __syncthreads();
#pragma unroll
    for (int i = 0; i < 8; ++i) {
      const int e = threadIdx.x + i * 256;   // 2048 elems
      const int r = e >> 7, c = e & 127;
      Vout[((size_t)tile * 16 + r) * GNN_D + c] = Vt[r * A_STRIDE + c] * rrow[r];
    }
    __syncthreads();
  }
}

// ------------------------------------------------------------------------
// Kernel D: out[b,:5] = [U[b], I[b]] @ Wout  — one wave per row, shuffle-reduce.
// ------------------------------------------------------------------------
__global__ __launch_bounds__(256) void gnn_out_proj(
    const float* __restrict__ U, const float* __restrict__ I,
    const float* __restrict__ Wout, float* __restrict__ out)
{
  const int lane = threadIdx.x & 31;
  const int wave = threadIdx.x >> 5;
  const int b = blockIdx.x * 8 + wave;
  if (b >= GNN_B) return;
  float o0 = 0.f, o1 = 0.f, o2 = 0.f, o3 = 0.f, o4 = 0.f;
#pragma unroll
  for (int i = 0; i < 4; ++i) {
    const int dd = lane + i * 32;
    const float u = U[(size_t)b * GNN_D + dd];
    const float* w = Wout + (size_t)dd * 5;
    o0 += u * w[0]; o1 += u * w[1]; o2 += u * w[2]; o3 += u * w[3]; o4 += u * w[4];
    const float it = I[(size_t)b * GNN_D + dd];
    const float* w2 = Wout + (size_t)(GNN_D + dd) * 5;
    o0 += it * w2[0]; o1 += it * w2[1]; o2 += it * w2[2]; o3 += it * w2[3]; o4 += it * w2[4];
  }
#pragma unroll
  for (int off = 16; off > 0; off >>= 1) {
    o0 += __shfl_down(o0, off);
    o1 += __shfl_down(o1, off);
    o2 += __shfl_down(o2, off);
    o3 += __shfl_down(o3, off);
    o4 += __shfl_down(o4, off);
  }
  if (lane == 0) {
    float* po = out + (size_t)b * 5;
    po[0] = o0; po[1] = o1; po[2] = o2; po[3] = o3; po[4] = o4;
  }
}

// ------------------------------------------------------------------------
extern "C" void kernel_launch(void* const* d_in, const int* in_sizes, int n_in,
                              void* d_out, int out_size, void* d_ws, size_t ws_size,
                              hipStream_t stream)
{
  (void)in_sizes; (void)n_in; (void)out_size; (void)ws_size;

  const int*   u_indices  = (const int*)d_in[0];
  const int*   v_indices  = (const int*)d_in[1];
  const int*   u_supports = (const int*)d_in[2];
  const int*   v_supports = (const int*)d_in[3];
  const int*   item_val   = (const int*)d_in[4];
  const int*   user_val   = (const int*)d_in[5];
  const float* u_features = (const float*)d_in[6];
  const float* v_features = (const float*)d_in[7];
  const float* Wu         = (const float*)d_in[8];
  const float* Wv         = (const float*)d_in[9];
  const float* Wout       = (const float*)d_in[10];
  const float* i_ew       = (const float*)d_in[11];
  const float* u_ew       = (const float*)d_in[12];
  const float* W_self_v   = (const float*)d_in[13];
  const float* W_neigh_v  = (const float*)d_in[14];
  const float* b_v        = (const float*)d_in[15];
  const float* W_self_u   = (const float*)d_in[16];
  const float* W_neigh_u  = (const float*)d_in[17];
  const float* b_u        = (const float*)d_in[18];

  float* ws = (float*)d_ws;
  float* Tu = ws;                               // [N,128]
  float* Tv = Tu + (size_t)GNN_N * GNN_D;       // [N,128]
  float* Hu = Tv + (size_t)GNN_N * GNN_D;       // [B,128] normalized user self
  float* Hi = Hu + (size_t)GNN_B * GNN_D;       // [B,128] normalized item self
  float* Mv = Hi + (size_t)GNN_B * GNN_D;       // [B,128] v-support neigh mean
  float* Mu = Mv + (size_t)GNN_B * GNN_D;       // [B,128] u-support neigh mean
  float* Uv = Mu + (size_t)GNN_B * GNN_D;       // [B,128] final user_vec
  float* Iv = Uv + (size_t)GNN_B * GNN_D;       // [B,128] final item_vec

  gnn_transform_relu<<<1024, 256, 0, stream>>>(u_features, Wu, Tu, GNN_N);
  gnn_transform_relu<<<1024, 256, 0, stream>>>(v_features, Wv, Tv, GNN_N);
  gnn_support_agg<<<GNN_B, 128, 0, stream>>>(u_indices, v_indices, u_supports,
                                             v_supports, item_val, user_val,
                                             Tu, Tv, i_ew, u_ew, Hu, Hi, Mv, Mu);
  gnn_convolve_norm<<<512, 256, 0, stream>>>(Hu, Mv, W_self_v, W_neigh_v, b_v, Uv);
  gnn_convolve_norm<<<512, 256, 0, stream>>>(Hi, Mu, W_self_u, W_neigh_u, b_u, Iv);
  gnn_out_proj<<<GNN_B / 8, 256, 0, stream>>>(Uv, Iv, Wout, (float*)d_out);
}